// FRAPModel_24824910970915
// MI455X (gfx1250) — compile-verified
//
#include <hip/hip_runtime.h>

typedef __attribute__((ext_vector_type(2))) float v2f;
typedef __attribute__((ext_vector_type(8))) float v8f;

// ---------------------------------------------------------------------------
// Kernel 1: fold the entire network into an affine map
//   out[b,s] = N[s,:] . ob_flat[b,:]  +  D[s],   N: 8x16, D: 8
// Parallelized over 64 threads in 3 stages to keep the critical path short
// (~450 VALU ops instead of ~2500 serial).  Writes 136 floats to d_ws.
// ---------------------------------------------------------------------------
__global__ __launch_bounds__(64) void frap_precompute(
    const int*   __restrict__ rel,
    const float* __restrict__ Wd, const float* __restrict__ bd,
    const float* __restrict__ W1, const float* __restrict__ b1,
    const float* __restrict__ Erel,
    const float* __restrict__ Wr, const float* __restrict__ br,
    const float* __restrict__ W2, const float* __restrict__ b2,
    const float* __restrict__ W3, const float* __restrict__ b3,
    float* __restrict__ nd)
{
    __shared__ float sv[20];      // v[c] = sum_o W3[o]*W2[o,c]
    __shared__ float sc0;         // W3.b2 + b3
    __shared__ float sA[56][8];   // folded per-(s,j) pair weights
    __shared__ float sB[56];      // folded per-(s,j) bias
    int t = threadIdx.x;

    // Stage 1: collapse W2/W3 into a 20-vector
    if (t < 20) {
        float a = 0.f;
        for (int o = 0; o < 20; ++o) a += W3[o] * W2[o * 20 + t];
        sv[t] = a;
    }
    if (t == 20) {
        float a = b3[0];
        for (int o = 0; o < 20; ++o) a += W3[o] * b2[o];
        sc0 = a;
    }
    __syncthreads();

    // Stage 2: one thread per (s,j) pair folds mask*W1 through v
    if (t < 56) {
        int r = rel[t];                       // relations[0, s, j]
        float A[8];
        for (int k = 0; k < 8; ++k) A[k] = 0.f;
        float Bb = 0.f;
        for (int o = 0; o < 20; ++o) {
            float mw = br[o];
            for (int c4 = 0; c4 < 4; ++c4) mw += Erel[r * 4 + c4] * Wr[o * 4 + c4];
            float m2 = sv[o] * mw;            // v[c] * mask[s,j,c]
            for (int k = 0; k < 8; ++k) A[k] += m2 * W1[o * 8 + k];
            Bb += m2 * b1[o];
        }
        for (int k = 0; k < 8; ++k) sA[t][k] = A[k];
        sB[t] = Bb;
    }
    __syncthreads();

    // Stage 3: one thread per output s assembles N[s,:], D[s]
    if (t < 8) {
        int s = t;
        float M[8][4];
        for (int p = 0; p < 8; ++p)
            for (int d = 0; d < 4; ++d) M[p][d] = 0.f;
        float Dacc = 7.f * sc0;
        for (int j = 0; j < 7; ++j) {
            int p   = (j < s) ? j : j + 1;    // K_IDX[s][j]
            int idx = s * 7 + j;
            Dacc += sB[idx];
            for (int d = 0; d < 4; ++d) {
                M[s][d] += sA[idx][d];        // src half of pair
                M[p][d] += sA[idx][4 + d];    // dst half of pair
            }
        }
        for (int p = 0; p < 8; ++p)
            for (int d = 0; d < 4; ++d) Dacc += M[p][d] * bd[d];
        for (int p = 0; p < 8; ++p)
            for (int c = 0; c < 2; ++c) {
                float a = 0.f;
                for (int d = 0; d < 4; ++d) a += M[p][d] * Wd[d * 2 + c];
                nd[s * 16 + p * 2 + c] = a;   // N[s, p*2+c]
            }
        nd[128 + s] = Dacc;                   // D[s]
    }
}

// ---------------------------------------------------------------------------
// Kernel 2: bandwidth-bound apply: out(B x 8) = ob(B x 16) @ N^T + D via
// V_WMMA_F32_16X16X4_F32.  B operand is zero-padded to 16 cols in LDS (no
// exec-masked loads), hoisted into registers once per wave; D folded into the
// accumulator init; grid-stride loop over 16-row tiles.
// ---------------------------------------------------------------------------
__global__ __launch_bounds__(256) void frap_apply(
    const float* __restrict__ ob,
    const float* __restrict__ nd,
    float* __restrict__ out,
    int nTiles)
{
    __shared__ float sNB[256];   // sNB[col*16 + k] = N[col][k], cols 8..15 = 0
    __shared__ float sD[16];     // D padded with zeros
    int tid = threadIdx.x;
    {
        int col = tid >> 4, k = tid & 15;
        sNB[tid] = (col < 8) ? nd[col * 16 + k] : 0.f;
        if (tid < 16) sD[tid] = (tid < 8) ? nd[128 + tid] : 0.f;
    }
    __syncthreads();

    int lane = tid & 31;
    int wave = tid >> 5;
    int m    = lane & 15;         // A-row / B-col / D-col index
    int hi   = lane >> 4;         // lane half
    int koff = hi * 2;            // lanes 16-31 hold K+2 (ISA 7.12.2)
    int col  = m;

    // Hoist the loop-invariant B operand (4x16 slices of N^T) into registers.
    v2f breg[4];
#pragma unroll
    for (int kk = 0; kk < 4; ++kk)
        breg[kk] = *(const v2f*)&sNB[col * 16 + kk * 4 + koff];
    float dadd = sD[col];

    int  gwave  = blockIdx.x * 8 + wave;
    int  nwaves = (int)gridDim.x * 8;
    bool valid  = col < 8;

    for (int tile = gwave; tile < nTiles; tile += nwaves) {   // wave-uniform
        long base = (long)tile * 16;
        const float* rp = ob + (base + m) * 16;

        v8f acc = {dadd, dadd, dadd, dadd, dadd, dadd, dadd, dadd};
#pragma unroll
        for (int kk = 0; kk < 4; ++kk) {
            v2f a = *(const v2f*)(rp + kk * 4 + koff);        // 16x4 A tile
            acc = __builtin_amdgcn_wmma_f32_16x16x4_f32(
                /*neg_a=*/false, a, /*neg_b=*/false, breg[kk],
                /*c_mod=*/(short)0, acc, /*reuse_a=*/false, /*reuse_b=*/false);
        }
        if (valid) {
            // D tile: lane half hi holds rows M = hi*8 + vv, column = col
            float* op = out + (base + hi * 8) * 8 + col;
#pragma unroll
            for (int vv = 0; vv < 8; ++vv)
                op[vv * 8] = acc[vv];
        }
    }
}

extern "C" void kernel_launch(void* const* d_in, const int* in_sizes, int n_in,
                              void* d_out, int out_size, void* d_ws, size_t ws_size,
                              hipStream_t stream) {
    const float* ob  = (const float*)d_in[0];
    const int*   rel = (const int*)  d_in[1];
    const float* Wd  = (const float*)d_in[2];
    const float* bd  = (const float*)d_in[3];
    const float* W1  = (const float*)d_in[4];
    const float* b1  = (const float*)d_in[5];
    const float* Er  = (const float*)d_in[6];
    const float* Wr  = (const float*)d_in[7];
    const float* br  = (const float*)d_in[8];
    const float* W2  = (const float*)d_in[9];
    const float* b2  = (const float*)d_in[10];
    const float* W3  = (const float*)d_in[11];
    const float* b3  = (const float*)d_in[12];
    float* out = (float*)d_out;
    float* nd  = (float*)d_ws;                // 136 floats of scratch

    int Btot   = in_sizes[0] / 16;            // 262144 batch rows
    int nTiles = (Btot + 15) / 16;            // 16384 16-row tiles

    // ~4 tiles per wave, 8 waves per block
    int waves  = (nTiles + 3) / 4;
    int blocks = (waves + 7) / 8;
    if (blocks < 1) blocks = 1;

    frap_precompute<<<1, 64, 0, stream>>>(rel, Wd, bd, W1, b1, Er, Wr, br,
                                          W2, b2, W3, b3, nd);
    frap_apply<<<blocks, 256, 0, stream>>>(ob, nd, out, nTiles);
}